// DFM_32255204393469
// MI455X (gfx1250) — compile-verified
//
#include <hip/hip_runtime.h>
#include <hip/hip_bf16.h>

typedef __attribute__((ext_vector_type(2))) float v2f;
typedef __attribute__((ext_vector_type(4))) float v4f;
typedef __attribute__((ext_vector_type(8))) float v8f;

#define BN_EPS 1e-3f
#define NORM_EPS 1e-12f

// ---------------------------------------------------------------------------
// WMMA f32 GEMM: C[m,n] = sum_k A(m,k)*B(k,n)
// TA: A(m,k)=A[k*lda+m] else A[m*lda+k]
// TB: B(k,n)=B[n*ldb+k] else B[k*ldb+n]
// EPI: 0 plain | 2 relu(bn(Y2)+acc) | 3 |acc|
// Wave computes a 32(M) x 32(N) tile = 4 V_WMMA_F32_16X16X4_F32 accumulators.
// k steps by 8: one b128 load per fragment covers two k-steps (lo/hi v2f),
// so each 8-k chunk is 4 vector loads : 8 WMMAs on the fast (TA=0,TB=1) path.
// Block = 8 waves (4 along M, 2 along N) = 128x64 tile. Kdim % 8 == 0.
// Out-of-range lanes load from clamped (valid) addresses; garbage only feeds
// output rows/cols the store guards never write -> K-loop is branch-free and
// EXEC stays all-1s as WMMA requires.
// ---------------------------------------------------------------------------
template <int TA, int TB, int EPI>
__global__ __launch_bounds__(256) void gemm_wmma(
    const float* __restrict__ A0, size_t sA, int lda,
    const float* __restrict__ B0, size_t sB, int ldb,
    float* __restrict__ C0, size_t sC, int ldc,
    int M, int Ncols, int Kdim,
    const float* __restrict__ Y2,
    const float* __restrict__ mean, const float* __restrict__ var,
    const float* __restrict__ gamma, const float* __restrict__ beta) {
  const int b = blockIdx.z;
  const float* A = A0 + (size_t)b * sA;
  const float* B = B0 + (size_t)b * sB;
  float* C = C0 + (size_t)b * sC;

  const int lane = threadIdx.x & 31;
  const int wave = threadIdx.x >> 5;
  const int wm = wave & 3;   // 4 waves along M
  const int wn = wave >> 2;  // 2 waves along N
  const int lo = lane & 15;
  const int hi = lane >> 4;

  const int tm = blockIdx.x * 128 + wm * 32;
  const int tn = blockIdx.y * 64 + wn * 32;

  const int rowA0 = tm + lo;
  const int rowA1 = rowA0 + 16;
  const int col0 = tn + lo;
  const int col1 = col0 + 16;
  // clamped addresses: safe loads; garbage only feeds never-stored outputs
  const int rowA0c = rowA0 < M ? rowA0 : M - 1;
  const int rowA1c = rowA1 < M ? rowA1 : M - 1;
  const int col0c = col0 < Ncols ? col0 : Ncols - 1;
  const int col1c = col1 < Ncols ? col1 : Ncols - 1;

  const float* pA0 = TA ? A + (size_t)(2 * hi) * lda + rowA0c
                        : A + (size_t)rowA0c * lda + 2 * hi;
  const float* pA1 = TA ? A + (size_t)(2 * hi) * lda + rowA1c
                        : A + (size_t)rowA1c * lda + 2 * hi;
  const float* pB0 = TB ? B + (size_t)col0c * ldb + 2 * hi
                        : B + (size_t)(2 * hi) * ldb + col0c;
  const float* pB1 = TB ? B + (size_t)col1c * ldb + 2 * hi
                        : B + (size_t)(2 * hi) * ldb + col1c;

  v8f acc[2][2];
#pragma unroll
  for (int mi = 0; mi < 2; ++mi)
#pragma unroll
    for (int ni = 0; ni < 2; ++ni)
#pragma unroll
      for (int i = 0; i < 8; ++i) acc[mi][ni][i] = 0.0f;

#pragma unroll 2
  for (int k = 0; k < Kdim; k += 8) {
    v4f a0, a1, b0, b1;
    if (TA) {
      a0.x = pA0[0];
      a0.y = pA0[lda];
      a0.z = pA0[(size_t)4 * lda];
      a0.w = pA0[(size_t)5 * lda];
      a1.x = pA1[0];
      a1.y = pA1[lda];
      a1.z = pA1[(size_t)4 * lda];
      a1.w = pA1[(size_t)5 * lda];
      pA0 += (size_t)8 * lda;
      pA1 += (size_t)8 * lda;
    } else {
      a0 = *(const v4f*)pA0;  // k-steps k and k+2 for this lane
      a1 = *(const v4f*)pA1;
      pA0 += 8;
      pA1 += 8;
    }
    if (TB) {
      b0 = *(const v4f*)pB0;
      b1 = *(const v4f*)pB1;
      pB0 += 8;
      pB1 += 8;
    } else {
      b0.x = pB0[0];
      b0.y = pB0[ldb];
      b0.z = pB0[(size_t)4 * ldb];
      b0.w = pB0[(size_t)5 * ldb];
      b1.x = pB1[0];
      b1.y = pB1[ldb];
      b1.z = pB1[(size_t)4 * ldb];
      b1.w = pB1[(size_t)5 * ldb];
      pB0 += (size_t)8 * ldb;
      pB1 += (size_t)8 * ldb;
    }
    const v2f a0l = __builtin_shufflevector(a0, a0, 0, 1);
    const v2f a0h = __builtin_shufflevector(a0, a0, 2, 3);
    const v2f a1l = __builtin_shufflevector(a1, a1, 0, 1);
    const v2f a1h = __builtin_shufflevector(a1, a1, 2, 3);
    const v2f b0l = __builtin_shufflevector(b0, b0, 0, 1);
    const v2f b0h = __builtin_shufflevector(b0, b0, 2, 3);
    const v2f b1l = __builtin_shufflevector(b1, b1, 0, 1);
    const v2f b1h = __builtin_shufflevector(b1, b1, 2, 3);
    acc[0][0] = __builtin_amdgcn_wmma_f32_16x16x4_f32(false, a0l, false, b0l,
                                                      (short)0, acc[0][0],
                                                      false, false);
    acc[0][1] = __builtin_amdgcn_wmma_f32_16x16x4_f32(false, a0l, false, b1l,
                                                      (short)0, acc[0][1],
                                                      false, false);
    acc[1][0] = __builtin_amdgcn_wmma_f32_16x16x4_f32(false, a1l, false, b0l,
                                                      (short)0, acc[1][0],
                                                      false, false);
    acc[1][1] = __builtin_amdgcn_wmma_f32_16x16x4_f32(false, a1l, false, b1l,
                                                      (short)0, acc[1][1],
                                                      false, false);
    acc[0][0] = __builtin_amdgcn_wmma_f32_16x16x4_f32(false, a0h, false, b0h,
                                                      (short)0, acc[0][0],
                                                      false, false);
    acc[0][1] = __builtin_amdgcn_wmma_f32_16x16x4_f32(false, a0h, false, b1h,
                                                      (short)0, acc[0][1],
                                                      false, false);
    acc[1][0] = __builtin_amdgcn_wmma_f32_16x16x4_f32(false, a1h, false, b0h,
                                                      (short)0, acc[1][0],
                                                      false, false);
    acc[1][1] = __builtin_amdgcn_wmma_f32_16x16x4_f32(false, a1h, false, b1h,
                                                      (short)0, acc[1][1],
                                                      false, false);
  }

  // epilogue (divergence allowed now; all WMMAs already issued)
#pragma unroll
  for (int mi = 0; mi < 2; ++mi) {
    const int mb = tm + mi * 16 + hi * 8;  // hi half-wave holds rows 8..15
#pragma unroll
    for (int ni = 0; ni < 2; ++ni) {
      const int n = tn + ni * 16 + lo;
      if (n >= Ncols) continue;
#pragma unroll
      for (int j = 0; j < 8; ++j) {
        const int m = mb + j;
        if (m >= M) break;
        float v = acc[mi][ni][j];
        if (EPI == 2) {
          const float y2 = Y2[(size_t)m * ldc + n];
          const float h =
              gamma[n] * (y2 - mean[n]) * rsqrtf(var[n] + BN_EPS) + beta[n];
          v = fmaxf(h + v, 0.0f);
        } else if (EPI == 3) {
          v = fabsf(v);
        }
        C[(size_t)m * ldc + n] = v;
      }
    }
  }
}

// Square-matrix transpose (weights -> contiguous GEMM B-fragment loads)
__global__ __launch_bounds__(256) void transpose_sq(const float* __restrict__ W,
                                                    float* __restrict__ Wt,
                                                    int D) {
  const int i = blockIdx.x * blockDim.x + threadIdx.x;
  if (i < D * D) {
    const int r = i / D;
    const int c = i - r * D;
    Wt[(size_t)c * D + r] = W[i];
  }
}

// Per-channel mean / population variance over all M rows; lanes map to
// consecutive channels -> coalesced.
__global__ __launch_bounds__(256) void bn_stats(const float* __restrict__ Y,
                                                int M, int D,
                                                float* __restrict__ mean,
                                                float* __restrict__ var) {
  const int ln = threadIdx.x & 31;
  const int rw = threadIdx.x >> 5;  // 0..7
  const int c = blockIdx.x * 32 + ln;
  float s = 0.0f, s2 = 0.0f;
  if (c < D) {
    for (int r = rw; r < M; r += 8) {
      const float v = Y[(size_t)r * D + c];
      s += v;
      s2 += v * v;
    }
  }
  __shared__ float red[8][33], red2[8][33];
  red[rw][ln] = s;
  red2[rw][ln] = s2;
  __syncthreads();
  if (rw == 0 && c < D) {
    float ts = 0.0f, ts2 = 0.0f;
#pragma unroll
    for (int i = 0; i < 8; ++i) {
      ts += red[i][ln];
      ts2 += red2[i][ln];
    }
    const float m = ts / (float)M;
    mean[c] = m;
    var[c] = ts2 / (float)M - m * m;
  }
}

// In-place H = relu(bn(Y))
__global__ __launch_bounds__(256) void bn_relu(float* __restrict__ Y,
                                               const float* __restrict__ mean,
                                               const float* __restrict__ var,
                                               const float* __restrict__ g,
                                               const float* __restrict__ b,
                                               size_t total, int D) {
  const size_t i = (size_t)blockIdx.x * blockDim.x + threadIdx.x;
  if (i >= total) return;
  const int c = (int)(i % (size_t)D);
  const float y = Y[i];
  const float h = g[c] * (y - mean[c]) * rsqrtf(var[c] + BN_EPS) + b[c];
  Y[i] = fmaxf(h, 0.0f);
}

// Per-batch Gauss-Jordan: solves A X = R in place (R becomes X). A = F F^T is
// SPD, so no pivoting. One block per batch.
__global__ __launch_bounds__(256) void gj_solve(float* __restrict__ Aall,
                                                float* __restrict__ Rall,
                                                int K) {
  float* A = Aall + (size_t)blockIdx.x * K * K;
  float* R = Rall + (size_t)blockIdx.x * K * K;
  __shared__ float rowA[128], rowR[128], colf[128];
  __shared__ float s_inv;
  const int t = threadIdx.x;
  for (int p = 0; p < K; ++p) {
    if (t == 0) s_inv = 1.0f / A[(size_t)p * K + p];
    __syncthreads();
    const float inv = s_inv;
    for (int j = t; j < K; j += blockDim.x) {
      const float av = A[(size_t)p * K + j] * inv;
      A[(size_t)p * K + j] = av;
      rowA[j] = av;
      const float rv = R[(size_t)p * K + j] * inv;
      R[(size_t)p * K + j] = rv;
      rowR[j] = rv;
    }
    if (t < K) colf[t] = (t == p) ? 0.0f : A[(size_t)t * K + p];
    __syncthreads();
    for (int w = t; w < K * K; w += blockDim.x) {
      const int r = w / K;
      const int j = w - r * K;
      const float f = colf[r];
      A[w] -= f * rowA[j];
      R[w] -= f * rowR[j];
    }
    __syncthreads();
  }
}

// C = X^T per batch, into the output tail.
__global__ __launch_bounds__(256) void transpose_C(const float* __restrict__ X,
                                                   float* __restrict__ Cout,
                                                   int K) {
  const int b = blockIdx.y;
  for (int w = blockIdx.x * blockDim.x + threadIdx.x; w < K * K;
       w += gridDim.x * blockDim.x) {
    const int k = w / K;
    const int l = w - k * K;
    Cout[(size_t)b * K * K + w] = X[(size_t)b * K * K + (size_t)l * K + k];
  }
}

// Column-normalize P (norm over row axis n, per column m), two passes.
__global__ __launch_bounds__(256) void colnorm(float* __restrict__ P, int N) {
  float* Pb = P + (size_t)blockIdx.z * N * N;
  const int ln = threadIdx.x & 31;
  const int col = blockIdx.x * 32 + ln;
  const int rw = threadIdx.x >> 5;
  float s = 0.0f;
  for (int r = rw; r < N; r += 8) {
    const float v = Pb[(size_t)r * N + col];
    s += v * v;
  }
  __shared__ float red[8][33];
  __shared__ float scale[32];
  red[rw][ln] = s;
  __syncthreads();
  if (rw == 0) {
    float t = 0.0f;
#pragma unroll
    for (int i = 0; i < 8; ++i) t += red[i][ln];
    scale[ln] = 1.0f / fmaxf(sqrtf(t), NORM_EPS);
  }
  __syncthreads();
  const float sc = scale[ln];
  for (int r = rw; r < N; r += 8) Pb[(size_t)r * N + col] *= sc;
}

// ---------------------------------------------------------------------------
extern "C" void kernel_launch(void* const* d_in, const int* in_sizes, int n_in,
                              void* d_out, int out_size, void* d_ws,
                              size_t ws_size, hipStream_t stream) {
  (void)in_sizes; (void)n_in; (void)out_size; (void)ws_size;
  const int B = 4, N = 4096, D = 352, K = 120;
  const int M = B * N;  // BN is over (B,N) jointly -> flatten to 16384 rows

  const float* src_desc = (const float*)d_in[0];
  const float* tar_desc = (const float*)d_in[1];
  const float* src_eig = (const float*)d_in[2];
  const float* tar_eig = (const float*)d_in[3];
  const float* W1 = (const float*)d_in[4];
  const float* g1 = (const float*)d_in[5];
  const float* b1 = (const float*)d_in[6];
  const float* W2 = (const float*)d_in[7];
  const float* g2 = (const float*)d_in[8];
  const float* b2 = (const float*)d_in[9];
  const float* W3 = (const float*)d_in[10];

  float* out = (float*)d_out;
  float* P = out;                         // [B,N,N]
  float* Cmat = out + (size_t)B * N * N;  // [B,K,K]

  // workspace carve-up (~81 MB)
  float* ws = (float*)d_ws;
  const size_t MD = (size_t)M * D;
  float* bufA = ws;          // H / Xnew (even iters)
  float* bufB = bufA + MD;   // H / Xnew (odd iters)
  float* bufY2 = bufB + MD;  // Y2
  float* Fm = bufY2 + MD;    // [B,K,D]
  float* Gm = Fm + (size_t)B * K * D;
  float* Am = Gm + (size_t)B * K * D;  // F F^T
  float* Xm = Am + (size_t)B * K * K;  // F G^T -> solution X
  float* Tm = Xm + (size_t)B * K * K;  // tar_eig @ X^T [B,N,K]
  float* meanv = Tm + (size_t)B * N * K;
  float* varv = meanv + D;
  float* W1t = varv + D;  // transposed weights
  float* W2t = W1t + (size_t)D * D;
  float* W3t = W2t + (size_t)D * D;

  const dim3 blk(256);
  const unsigned gT_w = (unsigned)((D * D + 255) / 256);
  transpose_sq<<<gT_w, blk, 0, stream>>>(W1, W1t, D);
  transpose_sq<<<gT_w, blk, 0, stream>>>(W2, W2t, D);
  transpose_sq<<<gT_w, blk, 0, stream>>>(W3, W3t, D);

  const dim3 gridMain((M + 127) / 128, (D + 63) / 64, 1);
  const dim3 gridStats((D + 31) / 32);

  for (int tensor = 0; tensor < 2; ++tensor) {
    const float* X = (tensor == 0) ? src_desc : tar_desc;
    for (int it = 0; it < 7; ++it) {
      float* H = (it & 1) ? bufB : bufA;
      // Y1 = X @ W1
      gemm_wmma<0, 1, 0><<<gridMain, blk, 0, stream>>>(
          X, 0, D, W1t, 0, D, H, 0, D, M, D, D, nullptr, nullptr, nullptr,
          nullptr, nullptr);
      bn_stats<<<gridStats, blk, 0, stream>>>(H, M, D, meanv, varv);
      bn_relu<<<(unsigned)((MD + 255) / 256), blk, 0, stream>>>(
          H, meanv, varv, g1, b1, MD, D);
      // Y2 = H @ W2
      gemm_wmma<0, 1, 0><<<gridMain, blk, 0, stream>>>(
          H, 0, D, W2t, 0, D, bufY2, 0, D, M, D, D, nullptr, nullptr, nullptr,
          nullptr, nullptr);
      bn_stats<<<gridStats, blk, 0, stream>>>(bufY2, M, D, meanv, varv);
      // Xnew = relu(bn(Y2) + X @ W3) fused epilogue, overwrite H
      gemm_wmma<0, 1, 2><<<gridMain, blk, 0, stream>>>(
          X, 0, D, W3t, 0, D, H, 0, D, M, D, D, bufY2, meanv, varv, g2, b2);
      X = H;
    }
    // F/G = eigen^T @ X_final : [B,K,D] (per-batch slices of flattened X)
    const float* eig = (tensor == 0) ? src_eig : tar_eig;
    float* FG = (tensor == 0) ? Fm : Gm;
    const dim3 gproj((K + 127) / 128, (D + 63) / 64, B);
    gemm_wmma<1, 0, 0><<<gproj, blk, 0, stream>>>(
        eig, (size_t)N * K, K, X, (size_t)N * D, D, FG, (size_t)K * D, D, K, D,
        N, nullptr, nullptr, nullptr, nullptr, nullptr);
  }

  // A = F F^T, Rhs = F G^T  [B,K,K]
  const dim3 gKK((K + 127) / 128, (K + 63) / 64, B);
  gemm_wmma<0, 1, 0><<<gKK, blk, 0, stream>>>(
      Fm, (size_t)K * D, D, Fm, (size_t)K * D, D, Am, (size_t)K * K, K, K, K,
      D, nullptr, nullptr, nullptr, nullptr, nullptr);
  gemm_wmma<0, 1, 0><<<gKK, blk, 0, stream>>>(
      Fm, (size_t)K * D, D, Gm, (size_t)K * D, D, Xm, (size_t)K * K, K, K, K,
      D, nullptr, nullptr, nullptr, nullptr, nullptr);

  // X = A^{-1} (F G^T); C = X^T
  gj_solve<<<B, blk, 0, stream>>>(Am, Xm, K);
  transpose_C<<<dim3(57, B), blk, 0, stream>>>(Xm, Cmat, K);

  // T = tar_eig @ X^T : [B,N,K]
  const dim3 gT((N + 127) / 128, (K + 63) / 64, B);
  gemm_wmma<0, 1, 0><<<gT, blk, 0, stream>>>(
      tar_eig, (size_t)N * K, K, Xm, (size_t)K * K, K, Tm, (size_t)N * K, K, N,
      K, K, nullptr, nullptr, nullptr, nullptr, nullptr);

  // P = | T @ src_eig^T | : [B,N,N]
  const dim3 gP((N + 127) / 128, (N + 63) / 64, B);
  gemm_wmma<0, 1, 3><<<gP, blk, 0, stream>>>(
      Tm, (size_t)N * K, K, src_eig, (size_t)N * K, K, P, (size_t)N * N, N, N,
      N, K, nullptr, nullptr, nullptr, nullptr, nullptr);

  // column-normalize P (norm over rows, per column)
  colnorm<<<dim3(N / 32, 1, B), blk, 0, stream>>>(P, N);
}